// ReviewLoss_1958505087535
// MI455X (gfx1250) — compile-verified
//
#include <hip/hip_runtime.h>
#include <math.h>

// ---------------------------------------------------------------------------
// ReviewLoss for MI455X (gfx1250):
//   Phase 1: per-row logsumexp CE, streamed at HBM rate via async DMA to LDS
//   Phase 2: exact radix-select of the k-th largest CE + thresholded mean
// ---------------------------------------------------------------------------

#define WAVE            32
#define WPB             8                  // waves per block (256 threads)
#define TILES           4                  // row-tiles of WPB rows per block
#define ROWS_PER_BLOCK  (WPB * TILES)      // 32 rows per block
#define MAXV            8                  // float4 chunks per lane (C <= 1024)
#define LDS_STRIDE      1008               // floats per LDS row slot (16B mult)
#define MAXJ            (LDS_STRIDE / 4)   // max float4 chunks stageable in LDS

// CDNA5 async global->LDS path (gfx1250). Guarded so both host pass and any
// toolchain lacking the builtins still compile (falls back to direct loads).
#if defined(__HIP_DEVICE_COMPILE__) && defined(__gfx1250__) && \
    __has_builtin(__builtin_amdgcn_global_load_async_to_lds_b128) && \
    __has_builtin(__builtin_amdgcn_s_wait_asynccnt)
#define USE_ASYNC_LDS 1
// Builtin prototype (from hipcc diagnostic): param0 = AS1 int4*, param1 = AS3
// int4*, then imm offset, imm cpol.
typedef int v4i __attribute__((vector_size(16)));
typedef __attribute__((address_space(1))) v4i g_v4i;
typedef __attribute__((address_space(3))) v4i l_v4i;
#endif

__device__ __forceinline__ int imin(int a, int b) { return a < b ? a : b; }

// Monotone float -> uint mapping (ascending order preserved)
__device__ __forceinline__ unsigned f2ord(float x) {
  unsigned u = __float_as_uint(x);
  return (u & 0x80000000u) ? ~u : (u | 0x80000000u);
}

// ---------------------------------------------------------------------------
// Phase 1: one wave per row. ce[row] = logsumexp(row) - row[target[row]]
// ---------------------------------------------------------------------------
__global__ __launch_bounds__(WPB * WAVE)
void ce_rows_kernel(const float* __restrict__ logits,
                    const int* __restrict__ tgt,
                    float* __restrict__ ce, int N, int C) {
  const int lane = threadIdx.x & (WAVE - 1);
  const int w    = threadIdx.x >> 5;
  // float4 path only when rows stay 16B aligned (C % 4 == 0); else scalar tail
  const int nvec   = (C & 3) ? 0 : (C >> 2);
  const int vecEnd = imin(nvec, imin(MAXJ, MAXV * WAVE));
  const bool fullIssue = (((vecEnd + WAVE - 1) >> 5) == MAXV);
  const long long base = (long long)blockIdx.x * ROWS_PER_BLOCK;

#if defined(USE_ASYNC_LDS)
  __shared__ __align__(16) float lds[2][WPB][LDS_STRIDE];
  {
    // Prologue: DMA tile 0 (this wave's row) into buffer 0.
    long long row = base + w;
    if (row < N) {
      const float* g = logits + row * (long long)C;
      float* l = &lds[0][w][0];
      for (int it = 0; it < MAXV; ++it) {
        int j = lane + it * WAVE;
        if (j < vecEnd) {
          __builtin_amdgcn_global_load_async_to_lds_b128(
              (g_v4i*)(g + 4 * j), (l_v4i*)(l + 4 * j), 0, 0);
        }
      }
    }
  }
#endif

  for (int t = 0; t < TILES; ++t) {
    const long long row = base + (long long)t * WPB + w;

#if defined(USE_ASYNC_LDS)
    bool prefetched = false;
    if (t + 1 < TILES) {
      long long nrow = base + (long long)(t + 1) * WPB + w;
      if (nrow < N) {
        const float* g = logits + nrow * (long long)C;
        float* l = &lds[(t + 1) & 1][w][0];
        for (int it = 0; it < MAXV; ++it) {
          int j = lane + it * WAVE;
          if (j < vecEnd) {
            __builtin_amdgcn_global_load_async_to_lds_b128(
                (g_v4i*)(g + 4 * j), (l_v4i*)(l + 4 * j), 0, 0);
          }
        }
        prefetched = true;
      }
    }
    // The newest MAXV async ops are the prefetch; waiting to <=MAXV guarantees
    // the current tile's DMA has landed in LDS while next tile stays in flight.
    if (prefetched && fullIssue) {
      __builtin_amdgcn_s_wait_asynccnt(MAXV);
    } else {
      __builtin_amdgcn_s_wait_asynccnt(0);
    }
#endif

    if (row < N) {
      const float* grow = logits + row * (long long)C;
#if defined(USE_ASYNC_LDS)
      const float4* src = (const float4*)&lds[t & 1][w][0];
#else
      const float4* src = (const float4*)grow;
#endif
      float4 v[MAXV];
      float m = -INFINITY;
#pragma unroll
      for (int it = 0; it < MAXV; ++it) {
        int j = lane + it * WAVE;
        if (j < vecEnd) {
          v[it] = src[j];
        } else {
          v[it] = make_float4(-INFINITY, -INFINITY, -INFINITY, -INFINITY);
        }
      }
#pragma unroll
      for (int it = 0; it < MAXV; ++it) {
        m = fmaxf(m, fmaxf(fmaxf(v[it].x, v[it].y), fmaxf(v[it].z, v[it].w)));
      }
      // scalar tail (C%4 != 0 or C > staged range) straight from global
      for (int c = 4 * vecEnd + lane; c < C; c += WAVE) m = fmaxf(m, grow[c]);
#pragma unroll
      for (int off = 16; off > 0; off >>= 1)
        m = fmaxf(m, __shfl_xor(m, off, WAVE));

      float s = 0.f;
#pragma unroll
      for (int it = 0; it < MAXV; ++it) {
        s += __expf(v[it].x - m) + __expf(v[it].y - m) +
             __expf(v[it].z - m) + __expf(v[it].w - m);
      }
      for (int c = 4 * vecEnd + lane; c < C; c += WAVE)
        s += __expf(grow[c] - m);
#pragma unroll
      for (int off = 16; off > 0; off >>= 1)
        s += __shfl_xor(s, off, WAVE);

      if (lane == 0) {
        int tc = tgt[row];
        float xt = grow[tc];              // single scalar, L2-hot
        ce[row] = __logf(s) + m - xt;     // -log_softmax at target
      }
    }
  }
}

// ---------------------------------------------------------------------------
// Phase 2: single block. Exact ascending rank-select via 4x8-bit radix
// histograms over ordered float keys (CE array is L2-resident), then mean of
// values >= lam over ALL N samples.
// ---------------------------------------------------------------------------
__global__ __launch_bounds__(1024)
void select_mean_kernel(const float* __restrict__ ce, float* __restrict__ out,
                        int N, int rank) {
  __shared__ unsigned hist[256];
  __shared__ unsigned s_prefix;
  __shared__ unsigned s_want;
  __shared__ double s_part[32];
  const int tid = threadIdx.x;

  if (tid == 0) { s_prefix = 0u; s_want = (unsigned)rank; }
  __syncthreads();

  for (int shift = 24; shift >= 0; shift -= 8) {
    if (tid < 256) hist[tid] = 0u;
    __syncthreads();
    const unsigned prefix = s_prefix;
    const unsigned mask = (shift == 24) ? 0u : (0xFFFFFFFFu << (shift + 8));
    for (int i = tid; i < N; i += (int)blockDim.x) {
      unsigned key = f2ord(ce[i]);
      if ((key & mask) == prefix)
        atomicAdd(&hist[(key >> shift) & 255u], 1u);
    }
    __syncthreads();
    if (tid == 0) {
      unsigned want = s_want, cum = 0u, b = 0u;
      for (; b < 256u; ++b) {
        unsigned c = hist[b];
        if (cum + c > want) break;
        cum += c;
      }
      s_want = want - cum;
      s_prefix = prefix | (b << shift);
    }
    __syncthreads();
  }

  const unsigned thresh = s_prefix;  // ordered key of lam (exact)
  double acc = 0.0;
  for (int i = tid; i < N; i += (int)blockDim.x) {
    float x = ce[i];
    if (f2ord(x) >= thresh) acc += (double)x;  // ce >= lam kept (ties kept)
  }
#pragma unroll
  for (int off = 16; off > 0; off >>= 1) acc += __shfl_xor(acc, off, WAVE);
  const int wid = tid >> 5, lane = tid & 31;
  if (lane == 0) s_part[wid] = acc;
  __syncthreads();
  if (tid == 0) {
    double tot = 0.0;
    for (int i = 0; i < 32; ++i) tot += s_part[i];
    out[0] = (float)(tot / (double)N);
  }
}

// ---------------------------------------------------------------------------
extern "C" void kernel_launch(void* const* d_in, const int* in_sizes, int n_in,
                              void* d_out, int out_size, void* d_ws,
                              size_t ws_size, hipStream_t stream) {
  const float* logits = (const float*)d_in[0];
  const int*   tgt    = (const int*)d_in[1];
  const int N = in_sizes[1];
  const int C = (N > 0) ? (in_sizes[0] / N) : 0;

  float* ce = (float*)d_ws;  // N floats of scratch (128 KB for N=32768)

  int k = (int)((double)N * 0.3);  // REVIEW_RATIO = 0.3 (module constant)
  int rank = N - 1 - k;            // ascending rank of sorted_desc[k]
  if (rank < 0) rank = 0;

  const int grid = (N + ROWS_PER_BLOCK - 1) / ROWS_PER_BLOCK;
  ce_rows_kernel<<<dim3(grid), dim3(WPB * WAVE), 0, stream>>>(logits, tgt, ce,
                                                              N, C);
  select_mean_kernel<<<dim3(1), dim3(1024), 0, stream>>>(ce, (float*)d_out, N,
                                                         rank);
}